// PartialPermutation_77936476553326
// MI455X (gfx1250) — compile-verified
//
#include <hip/hip_runtime.h>
#include <cstdint>

#define TPB  1024
#define LMAX 16384

typedef int v4i __attribute__((ext_vector_type(4)));
typedef __attribute__((address_space(1))) v4i g_v4i;   // global int4
typedef __attribute__((address_space(3))) v4i l_v4i;   // LDS int4
typedef __attribute__((address_space(3))) void as3_void;

// ---- CDNA5 async global->LDS copy (16 bytes per lane), tracked by ASYNCcnt ----
__device__ __forceinline__ void async_copy_b128(const float* __restrict__ g, float* l) {
#if __has_builtin(__builtin_amdgcn_global_load_async_to_lds_b128)
    __builtin_amdgcn_global_load_async_to_lds_b128((g_v4i*)g, (l_v4i*)l, 0, 0);
#else
    unsigned lds_off = (unsigned)(uintptr_t)(as3_void*)l;          // AS3 ptrtoint = LDS byte offset
    unsigned long long ga = (unsigned long long)(uintptr_t)g;
    asm volatile("global_load_async_to_lds_b128 %0, %1, off"
                 :: "v"(lds_off), "v"(ga) : "memory");
#endif
}

__device__ __forceinline__ void wait_async0() {
#if __has_builtin(__builtin_amdgcn_s_wait_asynccnt)
    __builtin_amdgcn_s_wait_asynccnt(0);
#else
    asm volatile("s_wait_asynccnt 0" ::: "memory");
#endif
}

// ---- cheap deterministic per-(row,pos,salt) uniform draw ----
__device__ __forceinline__ unsigned mix32(unsigned a) {
    a ^= a >> 16; a *= 0x7feb352dU;
    a ^= a >> 15; a *= 0x846ca68bU;
    a ^= a >> 16; return a;
}
__device__ __forceinline__ unsigned rnd32(unsigned row, unsigned pos, unsigned salt) {
    return mix32(row * 0x9E3779B9u ^ mix32(pos + salt * 0x85EBCA6Bu + 0x1234567u));
}

// ---- cycle-walking bijection on [0, cnt): O(1)-per-element random permutation ----
__device__ __forceinline__ unsigned permute_rank(unsigned r, unsigned cnt,
                                                 unsigned kbits, unsigned seed) {
    const unsigned mask = (1u << kbits) - 1u;
    const unsigned s1 = (kbits >> 1) + 1u;          // in [2, k] for k>=2
    const unsigned s2 = kbits - (kbits >> 1);       // in [1, k-1] for k>=2
    unsigned x = r & mask;
    do {
        x = (x * 0x2C1B3C6Du + seed) & mask;
        x ^= x >> s1;
        x = (x * 0x297A2D39u + (seed >> 5)) & mask;
        x ^= x >> s2;
    } while (x >= cnt);
    return x;
}

__global__ __launch_bounds__(TPB)
void PartialPermutation_kernel(const float* __restrict__ x,
                               const int*   __restrict__ real_len,
                               float*       __restrict__ out,
                               int L)
{
    __shared__ float    s_row[LMAX];    // 64 KB: staged input row
    __shared__ int      s_idx[LMAX];    // 64 KB: selected positions, ascending
    __shared__ unsigned s_mask[TPB];    // per-chunk selection bitmasks
    __shared__ int      s_scan[TPB];    // inclusive scan of per-chunk counts

    const int b     = blockIdx.x;
    const int tid   = threadIdx.x;
    const int bs    = blockDim.x;       // TPB
    const int items = L / bs;           // 16 for L=16384 (<=32 required)

    const float* xrow = x + (size_t)b * (size_t)L;

    // ---- Phase 1: kick off async DMA of the whole row into LDS (overlaps phase 2)
    const int nvec = L >> 2;            // float4 chunks
    for (int q = tid; q < nvec; q += bs)
        async_copy_b128(xrow + 4 * q, s_row + 4 * q);

    const int rl = real_len[b];

    // ---- Phase 2: Bernoulli selection + workgroup scan (+ rejection resample)
    unsigned m = 0;
    int total = 0;
    for (int attempt = 0; attempt < 8; ++attempt) {
        m = 0;
        for (int k = 0; k < items; ++k) {
            const int pos = tid * items + k;
            const bool eligible = (pos >= 1) && (pos <= rl - 2);
            if (eligible) {
                const unsigned r = rnd32((unsigned)b, (unsigned)pos, (unsigned)attempt);
                const float u = (float)(r >> 8) * (1.0f / 16777216.0f);
                if (u < 0.3f) m |= (1u << k);
            }
        }
        s_mask[tid] = m;
        s_scan[tid] = __popc(m);
        __syncthreads();
        // Hillis-Steele inclusive scan over 1024 chunk counts
        for (int off = 1; off < bs; off <<= 1) {
            const int v   = s_scan[tid];
            const int add = (tid >= off) ? s_scan[tid - off] : 0;
            __syncthreads();
            s_scan[tid] = v + add;
            __syncthreads();
        }
        total = s_scan[bs - 1];
        __syncthreads();                // protect s_scan/s_mask before a possible redraw
        if (total >= 2) break;          // uniform decision across the workgroup
    }

    // ---- Phase 3: compact selected positions (ascending) into s_idx
    if (total >= 2) {
        int base = s_scan[tid] - __popc(m);     // exclusive offset of this chunk
        for (int k = 0; k < items; ++k)
            if ((m >> k) & 1u) s_idx[base++] = tid * items + k;
    }

    // Drain this wave's async DMA, then barrier: row + index list visible to all.
    wait_async0();
    __syncthreads();

    // ---- Phase 4: permuted gather out of LDS, B128 coalesced stores
    const unsigned cnt = (unsigned)total;
    unsigned kbits = 2;
    if (cnt > 2) {
        kbits = 32u - (unsigned)__clz((int)(cnt - 1u));  // ceil(log2(cnt))
        if (kbits < 2) kbits = 2;
    }
    const unsigned seed = rnd32(0x5EED5EEDu, (unsigned)b, 0x77u) | 1u;

    float* orow = out + (size_t)b * (size_t)L;

    if ((items & 3) == 0) {
        // Vector path: 4 consecutive positions share one selection chunk.
        for (int q = tid; q < nvec; q += bs) {
            const int      pos0      = q << 2;
            const int      c         = pos0 / items;
            const int      btBase    = pos0 - c * items;
            const unsigned cm        = s_mask[c];
            const int      chunkExcl = s_scan[c] - __popc(cm);
            float4 v;
            float* vv = reinterpret_cast<float*>(&v);
#pragma unroll
            for (int e = 0; e < 4; ++e) {
                const int bt = btBase + e;
                if ((cnt >= 2u) && ((cm >> bt) & 1u)) {
                    const int r = chunkExcl + __popc(cm & ((1u << bt) - 1u));
                    const unsigned j = permute_rank((unsigned)r, cnt, kbits, seed);
                    vv[e] = s_row[s_idx[j]];   // LDS-served random gather
                } else {
                    vv[e] = s_row[pos0 + e];
                }
            }
            *reinterpret_cast<float4*>(orow + pos0) = v;   // global_store_b128
        }
    } else {
        // Scalar fallback for odd geometries.
        for (int pos = tid; pos < L; pos += bs) {
            const int      c  = pos / items;
            const int      bt = pos - c * items;
            const unsigned cm = s_mask[c];
            float v;
            if ((cnt >= 2u) && ((cm >> bt) & 1u)) {
                const int r = (s_scan[c] - __popc(cm)) + __popc(cm & ((1u << bt) - 1u));
                const unsigned j = permute_rank((unsigned)r, cnt, kbits, seed);
                v = s_row[s_idx[j]];
            } else {
                v = s_row[pos];
            }
            orow[pos] = v;
        }
    }
}

extern "C" void kernel_launch(void* const* d_in, const int* in_sizes, int n_in,
                              void* d_out, int out_size, void* d_ws, size_t ws_size,
                              hipStream_t stream) {
    (void)n_in; (void)d_ws; (void)ws_size; (void)out_size;
    const float* x  = (const float*)d_in[0];   // x: [B, L] float32
    const int*   rl = (const int*)d_in[1];     // real_len: [B] (int, per harness convention)
    float*       o  = (float*)d_out;

    const int B = in_sizes[1];
    int L = in_sizes[0] / (B > 0 ? B : 1);
    if (L > LMAX) L = LMAX;                    // sized for the reference shape (16384)

    PartialPermutation_kernel<<<dim3(B), dim3(TPB), 0, stream>>>(x, rl, o, L);
}